// FLEighConduit_33956011442441
// MI455X (gfx1250) — compile-verified
//
#include <hip/hip_runtime.h>
#include <hip/hip_bf16.h>

typedef __attribute__((ext_vector_type(2))) float v2f;
typedef __attribute__((ext_vector_type(8))) float v8f;

// ---- WMMA helpers: 16x16 f32 tiles, column-major in LDS ---------------------
// A-matrix 16x4 layout (ISA 7.12.2): lanes 0-15 hold M=0..15, VGPR v holds
// K = v + 2*(lane>=16). B 4x16 mirrored. C/D: lane holds col=lane&15, rows
// v + 8*(lane>=16).

__device__ __forceinline__ float tileElem(const float* t, int r, int c, bool trans) {
    return trans ? t[r * 16 + c] : t[c * 16 + r];
}

__device__ __forceinline__ v2f loadAfrag(const float* t, int kb, bool trans, int lane) {
    int m = lane & 15;
    int k0 = kb * 4 + ((lane >> 4) << 1);
    v2f a;
    a.x = tileElem(t, m, k0, trans);
    a.y = tileElem(t, m, k0 + 1, trans);
    return a;
}

__device__ __forceinline__ v2f loadBfrag(const float* t, int kb, int lane) {
    int n = lane & 15;
    int k0 = kb * 4 + ((lane >> 4) << 1);
    v2f b;
    b.x = t[n * 16 + k0];
    b.y = t[n * 16 + k0 + 1];
    return b;
}

__device__ __forceinline__ v8f mm16(const float* tA, const float* tB, bool transA, int lane) {
    v8f acc = {0.f, 0.f, 0.f, 0.f, 0.f, 0.f, 0.f, 0.f};
#pragma unroll
    for (int kb = 0; kb < 4; kb++) {
        acc = __builtin_amdgcn_wmma_f32_16x16x4_f32(
            false, loadAfrag(tA, kb, transA, lane),
            false, loadBfrag(tB, kb, lane),
            (short)0, acc, false, false);
    }
    return acc;
}

__device__ __forceinline__ void storeTile(float* t, v8f d, int lane) {
    int c = lane & 15, hi = lane >> 4;
#pragma unroll
    for (int v = 0; v < 8; v++) t[c * 16 + hi * 8 + v] = d[v];
}

// ---- main kernel: one batch per lane (phase 1/3), wave-coop WMMA (phase 2) --
__global__ void __launch_bounds__(32)
fleigh_kernel(const float* __restrict__ Ag, float* __restrict__ out, int Btot) {
    __shared__ float Ms[32][8][64];   // f32 snapshots of M_1..M_8 per batch
    __shared__ float Vb[32][64];      // V, column-major (col*8+row) per batch
    __shared__ float evB[32][8];      // Rayleigh quotients
    __shared__ float rsB[32];         // refinement residual
    __shared__ float tV[256];         // 16x16 tiles (col-major)
    __shared__ float tS[256];
    __shared__ float tT[256];

    const int lb   = threadIdx.x;           // local batch / lane
    const int lane = lb;
    const int hi   = lane >> 4;
    const int cl16 = lane & 15;
    const int gBase = blockIdx.x * 32;
    const int b = gBase + lb;

    // output slabs (reference return order, all f32)
    float* o_se = out;
    float* o_sv = out + (size_t)Btot * 8;
    float* o_cc = out + (size_t)Btot * 72;
    float* o_fr = out + (size_t)Btot * 80;
    float* o_st = out + (size_t)Btot * 88;
    float* o_eo = out + (size_t)Btot * 96;
    float* o_rr = out + (size_t)Btot * 104;

    // =================== Phase 1: per-lane scalar/f64 work ===================
    float a0[64];
#pragma unroll
    for (int i = 0; i < 64; i++) a0[i] = Ag[(size_t)b * 64 + i];

    float fs = 0.f;
#pragma unroll
    for (int i = 0; i < 64; i++) fs += a0[i] * a0[i];
    float scale = sqrtf(fs) * 0.35355339059327373f;   // /sqrt(8)
    scale = fmaxf(scale, 1e-12f);
    float An[64];
#pragma unroll
    for (int i = 0; i < 64; i++) An[i] = a0[i] / scale;

    // Faddeev-LeVerrier in f64 (registers), M_k snapshots to LDS in f32
    double c_[9];
#pragma unroll
    for (int i = 0; i < 9; i++) c_[i] = 0.0;
    c_[8] = 1.0;
    double Mk_[64];
#pragma unroll
    for (int i = 0; i < 64; i++) Mk_[i] = 0.0;

#pragma unroll
    for (int k = 1; k <= 8; k++) {
        double Mn[64];
#pragma unroll
        for (int i = 0; i < 8; i++) {
#pragma unroll
            for (int j = 0; j < 8; j++) {
                double s = 0.0;
#pragma unroll
                for (int t = 0; t < 8; t++) s += (double)An[i * 8 + t] * Mk_[t * 8 + j];
                if (i == j) s += c_[9 - k];
                Mn[i * 8 + j] = s;
            }
        }
        double tr = 0.0;
#pragma unroll
        for (int i = 0; i < 64; i++) {
            Ms[lb][k - 1][i] = (float)Mn[i];
            tr += (double)An[i] * Mn[i];
            Mk_[i] = Mn[i];
        }
        c_[8 - k] = -tr / (double)k;
    }

#pragma unroll
    for (int j = 0; j < 8; j++) o_cc[(size_t)b * 8 + j] = (float)c_[j];

    // initial guesses: sorted diagonal + linspace perturbation
    float dg[8];
#pragma unroll
    for (int i = 0; i < 8; i++) dg[i] = An[i * 9];
    for (int i = 1; i < 8; i++) {
        float e = dg[i];
        int j = i - 1;
        while (j >= 0 && dg[j] > e) { dg[j + 1] = dg[j]; j--; }
        dg[j + 1] = e;
    }
    double zi_[8];
    for (int r = 0; r < 8; r++)
        zi_[r] = (double)dg[r] + (-1.0e-4 + (double)r * (2.0e-4 / 7.0));

    // Laguerre with deflation (f64)
    double clp[9];
#pragma unroll
    for (int i = 0; i < 9; i++) clp[i] = c_[i];
    double roots_[8];
    float frA[8], stA[8];

    for (int ri = 0; ri < 8; ri++) {
        const int deg = 8 - ri;
        const double dd = (double)deg;
        double z = zi_[ri];
        float fr = 0.f, st = 5.f;
        for (int lag = 0; lag < 5; lag++) {
            double pv = clp[deg], dp = 0.0, d2 = 0.0;
            for (int j = deg - 1; j >= 0; j--) {
                d2 = d2 * z + dp;
                dp = dp * z + pv;
                pv = pv * z + clp[j];
            }
            float dpa = (float)fabs(dp);
            fr += 1.0f / (dpa + 1e-8f);
            float pva = (float)fabs(pv);
            if (pva < 1e-6f && st == 5.0f) st = (float)lag;
            bool ok = fabs(pv) > 1e-30;
            double ps = ok ? pv : 1.0;
            double G = ok ? dp / ps : 0.0;
            double H = G * G - (ok ? 2.0 * d2 / ps : 0.0);
            double disc = (dd - 1.0) * (dd * H - G * G);
            if (disc < 0.0) disc = 0.0;
            double sq = sqrt(disc);
            double gp = G + sq, gm = G - sq;
            double den = (fabs(gp) >= fabs(gm)) ? gp : gm;
            bool dok = fabs(den) > 1e-20;
            z -= dok ? dd / den : 0.0;
        }
        roots_[ri] = z;
        frA[ri] = fr;
        stA[ri] = st;
        // synthetic-division deflation
        double bb = clp[deg];
        double nc[9];
        for (int j = deg - 1; j >= 1; j--) {
            double bn = clp[j] + z * bb;
            nc[j] = bb;
            bb = bn;
        }
        nc[0] = bb;
        for (int j = 0; j < deg; j++) clp[j] = nc[j];
    }

    // Newton polish on full polynomial
    for (int it = 0; it < 3; it++) {
        for (int r = 0; r < 8; r++) {
            double z = roots_[r], pv = 1.0, dp = 0.0;
            for (int j = 7; j >= 0; j--) {
                dp = dp * z + pv;
                pv = pv * z + c_[j];
            }
            if (fabs(dp) > 1e-30) roots_[r] = z - pv / dp;
        }
    }

    // Resolvent Horner -> eigenvector candidates -> V columns (LDS, col-major)
    for (int ri = 0; ri < 8; ri++) {
        double lam = roots_[ri];
        double R_[64];
#pragma unroll
        for (int i = 0; i < 64; i++) R_[i] = (double)Ms[lb][0][i];
        for (int k = 2; k <= 8; k++) {
#pragma unroll
            for (int i = 0; i < 64; i++) R_[i] = R_[i] * lam + (double)Ms[lb][k - 1][i];
        }
        double bestn = -1.0;
        double vcol[8];
#pragma unroll
        for (int cc = 0; cc < 8; cc++) {
            double s2 = 0.0;
#pragma unroll
            for (int r = 0; r < 8; r++) {
                double e = R_[r * 8 + cc];
                s2 += e * e;
            }
            if (s2 > bestn) {
                bestn = s2;
#pragma unroll
                for (int r = 0; r < 8; r++) vcol[r] = R_[r * 8 + cc];
            }
        }
        double inv = 1.0 / (sqrt(bestn) + 1e-30);
        for (int r = 0; r < 8; r++) Vb[lb][ri * 8 + r] = (float)(vcol[r] * inv);
    }
    __syncthreads();

    // =========== Phase 2: wave-cooperative WMMA (2 batches / tile) ===========
    for (int p = 0; p < 16; p++) {
        const int b0l = 2 * p, b1l = 2 * p + 1;

        // block-diagonal V tile
#pragma unroll
        for (int v = 0; v < 8; v++) {
            int e = v * 32 + lane, r = e & 15, cc = e >> 4;
            float val = 0.f;
            if (r < 8 && cc < 8) val = Vb[b0l][cc * 8 + r];
            else if (r >= 8 && cc >= 8) val = Vb[b1l][(cc - 8) * 8 + (r - 8)];
            tV[e] = val;
        }
        __syncthreads();

        v8f Y = mm16(tV, tV, true, lane);           // Y = V^T V
        v8f T1, X1;
#pragma unroll
        for (int v = 0; v < 8; v++) {
            int r = v + hi * 8;
            T1[v] = ((r == cl16) ? 3.0f : 0.0f) - Y[v];
            X1[v] = 0.5f * T1[v];                   // X1 = 0.5*I*T1
        }
        storeTile(tS, T1, lane);
        storeTile(tT, Y, lane);
        __syncthreads();
        v8f Y1 = mm16(tS, tT, false, lane);         // Y1 = 0.5*T1*Y
        v8f T2;
#pragma unroll
        for (int v = 0; v < 8; v++) {
            Y1[v] *= 0.5f;
            int r = v + hi * 8;
            T2[v] = ((r == cl16) ? 3.0f : 0.0f) - Y1[v];
        }
        __syncthreads();
        storeTile(tS, X1, lane);
        storeTile(tT, T2, lane);
        __syncthreads();
        v8f X2 = mm16(tS, tT, false, lane);         // X2 = 0.5*X1*T2
#pragma unroll
        for (int v = 0; v < 8; v++) X2[v] *= 0.5f;
        __syncthreads();
        storeTile(tS, X2, lane);
        __syncthreads();
        v8f Vp = mm16(tV, tS, false, lane);         // V' = V*X2
        __syncthreads();
        storeTile(tT, Vp, lane);
        __syncthreads();
        v8f G = mm16(tT, tT, true, lane);           // V'^T V'

        // residual = ||V'^T V' - I||_F per batch
        bool inBlk = (hi == 0 && cl16 < 8) || (hi == 1 && cl16 >= 8);
        float pr = 0.f;
        if (inBlk) {
#pragma unroll
            for (int v = 0; v < 8; v++) {
                int r = v + hi * 8;
                float d = G[v] - ((r == cl16) ? 1.0f : 0.0f);
                pr += d * d;
            }
        }
        pr += __shfl_xor(pr, 1);
        pr += __shfl_xor(pr, 2);
        pr += __shfl_xor(pr, 4);
        if (lane == 0)  rsB[b0l] = sqrtf(pr);
        if (lane == 24) rsB[b1l] = sqrtf(pr);

        // AV with original (unscaled) A
#pragma unroll
        for (int v = 0; v < 8; v++) {
            int e = v * 32 + lane, r = e & 15, cc = e >> 4;
            float val = 0.f;
            if (r < 8 && cc < 8)
                val = Ag[((size_t)(gBase + b0l)) * 64 + r * 8 + cc];
            else if (r >= 8 && cc >= 8)
                val = Ag[((size_t)(gBase + b1l)) * 64 + (r - 8) * 8 + (cc - 8)];
            tS[e] = val;
        }
        __syncthreads();
        v8f AV = mm16(tS, tT, false, lane);         // A*V'

        float es = 0.f;
#pragma unroll
        for (int v = 0; v < 8; v++) es += Vp[v] * AV[v];
        if (hi == 0 && cl16 < 8)  evB[b0l][cl16] = es;         // column sums
        if (hi == 1 && cl16 >= 8) evB[b1l][cl16 - 8] = es;

        // write back V'
        if (hi == 0 && cl16 < 8) {
#pragma unroll
            for (int v = 0; v < 8; v++) Vb[b0l][cl16 * 8 + v] = Vp[v];
        }
        if (hi == 1 && cl16 >= 8) {
#pragma unroll
            for (int v = 0; v < 8; v++) Vb[b1l][(cl16 - 8) * 8 + v] = Vp[v];
        }
        __syncthreads();
    }

    // =================== Phase 3: sort, sign-fix, outputs ====================
    float ev[8];
    int idx[8];
#pragma unroll
    for (int i = 0; i < 8; i++) { ev[i] = evB[lb][i]; idx[i] = i; }
    for (int i = 1; i < 8; i++) {
        float e = ev[i];
        int id = idx[i];
        int j = i - 1;
        while (j >= 0 && ev[j] > e) {
            ev[j + 1] = ev[j];
            idx[j + 1] = idx[j];
            j--;
        }
        ev[j + 1] = e;
        idx[j + 1] = id;
    }

    for (int j = 0; j < 8; j++) {
        int pj = idx[j];
        float col[8];
        for (int r = 0; r < 8; r++) col[r] = Vb[lb][pj * 8 + r];
        float mv = -1.f;
        int mr = 0;
        for (int r = 0; r < 8; r++) {
            float av = fabsf(col[r]);
            if (av > mv) { mv = av; mr = r; }
        }
        float pivot = col[mr];
        float sgn = (float)((pivot > 0.f) - (pivot < 0.f));
        o_se[(size_t)b * 8 + j] = ev[j];
        o_fr[(size_t)b * 8 + j] = frA[pj];
        o_st[(size_t)b * 8 + j] = stA[pj];
        o_eo[(size_t)b * 8 + j] = (float)j;
        for (int r = 0; r < 8; r++)
            o_sv[(size_t)b * 64 + r * 8 + j] = col[r] * sgn;
    }
    o_rr[b] = rsB[lb];
}

extern "C" void kernel_launch(void* const* d_in, const int* in_sizes, int n_in,
                              void* d_out, int out_size, void* d_ws, size_t ws_size,
                              hipStream_t stream) {
    const float* A = (const float*)d_in[0];
    float* out = (float*)d_out;
    const int Btot = in_sizes[0] / 64;          // 65536
    dim3 grid(Btot / 32), block(32);
    hipLaunchKernelGGL(fleigh_kernel, grid, block, 0, stream, A, out, Btot);
}